// SAAMv3CLS_19310172963078
// MI455X (gfx1250) — compile-verified
//
#include <hip/hip_runtime.h>
#include <hip/hip_bf16.h>

#define B_    32
#define L_    4096
#define D_    768
#define S_    128
#define NASP  6
#define NRAT  5
#define NSEG  (B_ * S_)      // 4096 sentences
#define LN_EPS 1e-5f

typedef __attribute__((ext_vector_type(2))) float v2f;
typedef __attribute__((ext_vector_type(8))) float v8f;

// ---------------------------------------------------------------------------
// Kernel 0: find the 128 mark positions per batch row (block scan over L=4096)
// ---------------------------------------------------------------------------
__global__ __launch_bounds__(256) void marks_kernel(const int* __restrict__ p,
                                                    int* __restrict__ marks) {
    __shared__ int sh[256];
    const int b = blockIdx.x, tid = threadIdx.x;
    const int base = b * L_ + tid * 16;
    int loc[16];
    int c = 0;
#pragma unroll
    for (int i = 0; i < 16; ++i) { loc[i] = p[base + i]; c += loc[i]; }
    sh[tid] = c;
    __syncthreads();
    // Hillis-Steele inclusive scan
    for (int off = 1; off < 256; off <<= 1) {
        int v = (tid >= off) ? sh[tid - off] : 0;
        __syncthreads();
        sh[tid] += v;
        __syncthreads();
    }
    int rank = sh[tid] - c;  // exclusive prefix
#pragma unroll
    for (int i = 0; i < 16; ++i)
        if (loc[i]) marks[b * S_ + (rank++)] = tid * 16 + i;
}

// ---------------------------------------------------------------------------
// Kernel 1: segment means for both big tensors (the 805MB streaming pass)
// Segment s of batch b: tokens in (m[s-1], m[s]) exclusive (marks excluded).
// ---------------------------------------------------------------------------
__global__ __launch_bounds__(256) void segmean_kernel(const float* __restrict__ A,
                                                      const float* __restrict__ Sn,
                                                      const int* __restrict__ marks,
                                                      float* __restrict__ pa,
                                                      float* __restrict__ ps) {
    const int n = blockIdx.x, tid = threadIdx.x;
    const int b = n >> 7, s = n & (S_ - 1);
    const int end   = marks[b * S_ + s];
    const int start = (s == 0) ? 0 : (marks[b * S_ + s - 1] + 1);
    float a0 = 0.f, a1 = 0.f, a2 = 0.f, s0 = 0.f, s1 = 0.f, s2 = 0.f;
    size_t rb = ((size_t)b * L_ + (size_t)start) * D_ + tid;
    for (int t = start; t < end; ++t, rb += D_) {
        const float* pA = A + rb;
        const float* pS = Sn + rb;
        __builtin_prefetch(pA + D_, 0, 1);   // -> global_prefetch_b8
        __builtin_prefetch(pS + D_, 0, 1);
        float x0 = pA[0], x1 = pA[256], x2 = pA[512];
        float y0 = pS[0], y1 = pS[256], y2 = pS[512];
        a0 += x0; a1 += x1; a2 += x2;
        s0 += y0; s1 += y1; s2 += y2;
    }
    const float inv = 1.0f / (float)(end - start);
    size_t ob = (size_t)n * D_ + tid;
    pa[ob]       = a0 * inv; pa[ob + 256] = a1 * inv; pa[ob + 512] = a2 * inv;
    ps[ob]       = s0 * inv; ps[ob + 256] = s1 * inv; ps[ob + 512] = s2 * inv;
}

// ---------------------------------------------------------------------------
// Kernel 2: LayerNorm(pooled_a) @ W_asp via V_WMMA_F32_16X16X4_F32, softmax(7)
// One block = 16 sentences. 8 waves each cover 96 K-dims (24 WMMAs), partial
// 16x16 tiles reduced through LDS, then per-row softmax.
// W is stored transposed & zero-padded in LDS so both A and B fragments are
// unconditional contiguous ds_load_b64 (no exec-mask predication in the loop).
// ---------------------------------------------------------------------------
__global__ __launch_bounds__(256) void aspect_kernel(const float* __restrict__ pa,
                                                     const float* __restrict__ Wasp,
                                                     const float* __restrict__ lnsc,
                                                     const float* __restrict__ lnbi,
                                                     float* __restrict__ dist,
                                                     float* __restrict__ aspw) {
    __shared__ float shH[16][772];   // padded: stride 772 % 64 == 4 -> conflict-free
    __shared__ float shWt[16][772];  // W^T, cols 7..15 zero-padded
    __shared__ float shC[8][256];
    __shared__ float shL[256];
    __shared__ float shScale[D_];
    __shared__ float shBias[D_];

    const int tid = threadIdx.x;
    const int n0  = blockIdx.x * 16;

    // W^T with zero padding: shWt[n][k] = (n<7) ? W[k][n] : 0
#pragma unroll
    for (int n = 0; n < 16; ++n)
        for (int k = tid; k < D_; k += 256)
            shWt[n][k] = (n < 7) ? Wasp[k * 7 + n] : 0.f;
    for (int i = tid; i < D_; i += 256) { shScale[i] = lnsc[i]; shBias[i] = lnbi[i]; }
    for (int m = 0; m < 16; ++m) {
        size_t base = (size_t)(n0 + m) * D_ + tid;
        shH[m][tid]       = pa[base];
        shH[m][tid + 256] = pa[base + 256];
        shH[m][tid + 512] = pa[base + 512];
    }
    __syncthreads();

    // --- per-row LayerNorm: 16 threads per row, shuffle reduce within 16 lanes
    const int r = tid >> 4, lane16 = tid & 15;
    float sum = 0.f, sq = 0.f;
    for (int i = 0; i < 48; ++i) {
        float x = shH[r][lane16 + 16 * i];
        sum += x; sq += x * x;
    }
    for (int off = 8; off >= 1; off >>= 1) {
        sum += __shfl_xor(sum, off, 16);
        sq  += __shfl_xor(sq,  off, 16);
    }
    const float mean = sum * (1.0f / D_);
    const float var  = sq * (1.0f / D_) - mean * mean;
    const float rs   = rsqrtf(var + LN_EPS);
    for (int i = 0; i < 48; ++i) {
        int d = lane16 + 16 * i;
        float x = shH[r][d];
        shH[r][d] = (x - mean) * rs * shScale[d] + shBias[d];
    }
    __syncthreads();

    // --- WMMA: D(16x16) += A(16x4) * B(4x16), 24 steps per wave
    const int wave = tid >> 5, lane = tid & 31;
    const int half = lane >> 4, mn = lane & 15;
    v8f c = {};
    for (int i = 0; i < 24; ++i) {
        const int k0 = wave * 96 + i * 4;
        const int ka = k0 + 2 * half;
        v2f a, bb;
        a.x  = shH[mn][ka];        // pairs -> ds_load_b64
        a.y  = shH[mn][ka + 1];
        bb.x = shWt[mn][ka];       // pairs -> ds_load_b64, no predication
        bb.y = shWt[mn][ka + 1];
        c = __builtin_amdgcn_wmma_f32_16x16x4_f32(false, a, false, bb,
                                                  (short)0, c, false, false);
    }
#pragma unroll
    for (int i = 0; i < 8; ++i)
        shC[wave][(i + 8 * half) * 16 + mn] = c[i];   // C layout: M=i+8*half, N=mn
    __syncthreads();

    float logit = 0.f;
#pragma unroll
    for (int w = 0; w < 8; ++w) logit += shC[w][tid];
    shL[tid] = logit;   // shL[M*16 + N]
    __syncthreads();

    if (tid < 16) {
        float l[7], mx = -1e30f;
        for (int j = 0; j < 7; ++j) { l[j] = shL[tid * 16 + j]; mx = fmaxf(mx, l[j]); }
        float se = 0.f;
        for (int j = 0; j < 7; ++j) { l[j] = expf(l[j] - mx); se += l[j]; }
        const float inv = 1.0f / se;
        const size_t nr = (size_t)(n0 + tid);
        for (int j = 0; j < 7; ++j) dist[nr * 7 + j] = l[j] * inv;
        for (int j = 0; j < 6; ++j) aspw[nr * 6 + j] = l[j] * inv;
    }
}

// ---------------------------------------------------------------------------
// Kernel 3: per-batch weighted doc embedding, LayerNorm, @ W_sent (768x5)
// ---------------------------------------------------------------------------
__global__ __launch_bounds__(256) void doc_kernel(const float* __restrict__ ps,
                                                  const float* __restrict__ aspw,
                                                  const float* __restrict__ lnsc,
                                                  const float* __restrict__ lnbi,
                                                  const float* __restrict__ Wsent,
                                                  float* __restrict__ result) {
    __shared__ float red1[256];
    __shared__ float red2[256];
    __shared__ float bcast[2];
    __shared__ float pbuf[30 * 257];      // padded stride to dodge bank conflicts

    const int b = blockIdx.x, tid = threadIdx.x;
    float acc[6][3];
    float wsum[6];
#pragma unroll
    for (int a = 0; a < 6; ++a) { wsum[a] = 0.f; acc[a][0] = acc[a][1] = acc[a][2] = 0.f; }

    for (int s = 0; s < S_; ++s) {
        const int n = b * S_ + s;
        float w[6];
#pragma unroll
        for (int a = 0; a < 6; ++a) w[a] = aspw[n * 6 + a];
        const size_t base = (size_t)n * D_ + tid;
        const float p0 = ps[base], p1 = ps[base + 256], p2 = ps[base + 512];
#pragma unroll
        for (int a = 0; a < 6; ++a) {
            acc[a][0] = fmaf(w[a], p0, acc[a][0]);
            acc[a][1] = fmaf(w[a], p1, acc[a][1]);
            acc[a][2] = fmaf(w[a], p2, acc[a][2]);
            wsum[a] += w[a];
        }
    }

    float sc[3], bi[3], wr[3][5];
#pragma unroll
    for (int j = 0; j < 3; ++j) {
        const int d = tid + 256 * j;
        sc[j] = lnsc[d]; bi[j] = lnbi[d];
#pragma unroll
        for (int r2 = 0; r2 < 5; ++r2) wr[j][r2] = Wsent[d * 5 + r2];
    }

    float part[6][5];
    for (int a = 0; a < 6; ++a) {
        const float inv = 1.0f / wsum[a];
        const float e0 = acc[a][0] * inv, e1 = acc[a][1] * inv, e2 = acc[a][2] * inv;
        red1[tid] = e0 + e1 + e2;
        red2[tid] = e0 * e0 + e1 * e1 + e2 * e2;
        __syncthreads();
        for (int off = 128; off >= 1; off >>= 1) {
            if (tid < off) { red1[tid] += red1[tid + off]; red2[tid] += red2[tid + off]; }
            __syncthreads();
        }
        if (tid == 0) {
            const float mean = red1[0] * (1.0f / D_);
            const float var  = red2[0] * (1.0f / D_) - mean * mean;
            bcast[0] = mean;
            bcast[1] = rsqrtf(var + LN_EPS);
        }
        __syncthreads();
        const float mean = bcast[0], rs = bcast[1];
        const float e[3] = {e0, e1, e2};
        float pr[5] = {0.f, 0.f, 0.f, 0.f, 0.f};
#pragma unroll
        for (int j = 0; j < 3; ++j) {
            const float h = (e[j] - mean) * rs * sc[j] + bi[j];
#pragma unroll
            for (int r2 = 0; r2 < 5; ++r2) pr[r2] = fmaf(h, wr[j][r2], pr[r2]);
        }
#pragma unroll
        for (int r2 = 0; r2 < 5; ++r2) part[a][r2] = pr[r2];
        __syncthreads();   // protect red1/red2 before next aspect
    }

#pragma unroll
    for (int a = 0; a < 6; ++a)
#pragma unroll
        for (int r2 = 0; r2 < 5; ++r2)
            pbuf[(a * 5 + r2) * 257 + tid] = part[a][r2];
    __syncthreads();
    if (tid < 30) {
        float s = 0.f;
        for (int i = 0; i < 256; ++i) s += pbuf[tid * 257 + i];   // fixed order: deterministic
        result[b * 30 + tid] = s;
    }
}

// ---------------------------------------------------------------------------
extern "C" void kernel_launch(void* const* d_in, const int* in_sizes, int n_in,
                              void* d_out, int out_size, void* d_ws, size_t ws_size,
                              hipStream_t stream) {
    const float* aspect = (const float*)d_in[0];   // (B,L,D) f32
    const float* senti  = (const float*)d_in[1];   // (B,L,D) f32
    const int*   pidx   = (const int*)d_in[2];     // (B,L) i32
    // d_in[3] = n_sent scalar (==128, compile-time S_)
    const float* ln_a_s = (const float*)d_in[4];
    const float* ln_a_b = (const float*)d_in[5];
    const float* Wasp   = (const float*)d_in[6];   // (768,7)
    const float* ln_s_s = (const float*)d_in[7];
    const float* ln_s_b = (const float*)d_in[8];
    const float* Wsent  = (const float*)d_in[9];   // (768,5)

    float* result = (float*)d_out;                     // (32,6,5) = 960 floats
    float* dist   = result + B_ * NASP * NRAT;         // (4096,7) = 28672 floats

    // workspace layout
    int*   marks    = (int*)d_ws;                                  // 4096 ints
    float* pooled_a = (float*)((char*)d_ws + 16384);               // 4096x768
    float* pooled_s = pooled_a + (size_t)NSEG * D_;                // 4096x768
    float* aspw     = pooled_s + (size_t)NSEG * D_;                // 4096x6

    marks_kernel  <<<B_,        256, 0, stream>>>(pidx, marks);
    segmean_kernel<<<NSEG,      256, 0, stream>>>(aspect, senti, marks, pooled_a, pooled_s);
    aspect_kernel <<<NSEG / 16, 256, 0, stream>>>(pooled_a, Wasp, ln_a_s, ln_a_b, dist, aspw);
    doc_kernel    <<<B_,        256, 0, stream>>>(pooled_s, aspw, ln_s_s, ln_s_b, Wsent, result);
}